// Memory_10368051052717
// MI455X (gfx1250) — compile-verified
//
#include <hip/hip_runtime.h>
#include <stdint.h>

typedef __attribute__((ext_vector_type(2))) float v2f;
typedef __attribute__((ext_vector_type(8))) float v8f;
typedef __attribute__((ext_vector_type(4))) unsigned int v4u;
typedef __attribute__((ext_vector_type(8))) int v8i;
typedef __attribute__((ext_vector_type(4))) int v4i;

#define K_TOP     10
#define C_DIM     512
#define N_ITEMS   2000
#define HW        1024          // 32*32 spatial
#define ROWS_WG   128           // rows per workgroup (8 waves x 16)
#define KC        64            // K chunk staged in LDS
#define NCHUNK    (C_DIM / KC)  // 8
#define NTILE     (N_ITEMS / 16)// 125
#define TCHUNKS   (NTILE * NCHUNK) // 1000
#define QPAD      (KC + 4)      // padded LDS row stride (68 floats)
#define NEG_INF   (-3.0e38f)

__device__ __forceinline__ void topk_insert(float (&tv)[K_TOP], int (&ti)[K_TOP],
                                            float v, int gi) {
    if (v > tv[K_TOP - 1]) {
        tv[K_TOP - 1] = v; ti[K_TOP - 1] = gi;
#pragma unroll
        for (int q = K_TOP - 1; q > 0; --q) {
            if (tv[q] > tv[q - 1]) {
                float tf = tv[q]; tv[q] = tv[q - 1]; tv[q - 1] = tf;
                int   tn = ti[q]; ti[q] = ti[q - 1]; ti[q - 1] = tn;
            }
        }
    }
}

// TDM: async-load one mempool tile chunk [16 rows x 64 f32] (row stride 512 f32)
// into LDS with 4-DWORD padding after every 64 DWORDs -> LDS row stride 68 f32.
__device__ __forceinline__ void tdm_load_mem_tile(const float* gsrc, unsigned lds_byte) {
    uint64_t ga = (uint64_t)(uintptr_t)gsrc;
    v4u g0;
    g0[0] = 1u;                                    // count=1 (valid descriptor)
    g0[1] = lds_byte;                              // lds_addr
    g0[2] = (unsigned)(ga & 0xFFFFFFFFu);          // global_addr[31:0]
    g0[3] = (unsigned)((ga >> 32) & 0x1FFFFFFu)    // global_addr[56:32]
          | (2u << 30);                            // type = 2 ("image")
    v8i g1;
    g1[0] = (2 << 16)      // data_size = 4B
          | (1 << 20)      // pad_enable
          | (5 << 22)      // pad_interval: 64 DWORDs
          | (3 << 25);     // pad_amount: 4 DWORDs
    g1[1] = (int)(512u << 16);    // tensor_dim0 = 512 (low 16 @ bits 63:48)
    g1[2] = (int)(2000u << 16);   // tensor_dim1 = 2000 (low 16 @ bits 95:80)
    g1[3] = (int)(64u << 16);     // tile_dim0 = 64 elements
    g1[4] = 16;                   // tile_dim1 = 16 rows, tile_dim2 = 0
    g1[5] = 512;                  // tensor_dim0_stride = 512 elements
    g1[6] = 0; g1[7] = 0;
    v4i z4 = {0, 0, 0, 0};
#if defined(__clang_major__) && (__clang_major__ >= 23)
    v8i z8 = {0, 0, 0, 0, 0, 0, 0, 0};
    __builtin_amdgcn_tensor_load_to_lds(g0, g1, z4, z4, z8, 0);
#else
    __builtin_amdgcn_tensor_load_to_lds(g0, g1, z4, z4, 0);
#endif
}

// Zero-fill the dense attvec*att output region (262 MB write = HBM floor).
__global__ void zero_f4(float4* __restrict__ p, long long n4) {
    long long i = (long long)blockIdx.x * blockDim.x + threadIdx.x;
    long long stride = (long long)gridDim.x * blockDim.x;
    float4 z = {0.f, 0.f, 0.f, 0.f};
    for (; i < n4; i += stride) p[i] = z;
}

// Fused: att GEMM (fp32 WMMA, TDM-staged B) + top-10 + softmax + scatter + out.
__global__ __launch_bounds__(256) void memnet_fused(
    const float* __restrict__ x,        // [32,512,32,32] NCHW
    const float* __restrict__ mempool,  // [2000,512]
    float* __restrict__ out,            // [32,512,32,32] NCHW
    float* __restrict__ attout)         // [32768,2000] dense (pre-zeroed)
{
    __shared__ float qS[ROWS_WG][QPAD];        // query K-chunk (reg-pipelined)
    __shared__ float bS[2][16][QPAD];          // mempool tile chunk (TDM double-buffered)
    __shared__ float attS[8][16][17];          // per-wave 16x16 att tile scratch
    __shared__ float wS[8][16][K_TOP];         // softmax weights per row
    __shared__ int   idS[8][16][K_TOP];        // top-k indices per row

    const int tid  = threadIdx.x;
    const int wave = tid >> 5;
    const int lane = tid & 31;
    const int mrow = lane & 15;                // M (A/D) or N (B) index within tile
    const int hf   = lane >> 4;                // lane half -> K-pair / col-half selector

    const int rowBase = blockIdx.x * ROWS_WG;
    const int b       = rowBase >> 10;         // 128 | 1024: no batch crossing
    const int hwBase  = rowBase & (HW - 1);
    const float* xB   = x + (size_t)b * C_DIM * HW + hwBase;

    // ---- prologue: stage chunk 0 (qS direct, bS[0] via TDM)
    for (int i = tid; i < (ROWS_WG / 4) * KC; i += 256) {
        int r4 = (i & 31) << 2;
        int c  = i >> 5;
        const float4 v = *(const float4*)(xB + (size_t)c * HW + r4);
        qS[r4 + 0][c] = v.x; qS[r4 + 1][c] = v.y;
        qS[r4 + 2][c] = v.z; qS[r4 + 3][c] = v.w;
    }
    if (wave == 0) {
        tdm_load_mem_tile(mempool, (unsigned)(uintptr_t)&bS[0][0][0]);
        __builtin_amdgcn_s_wait_tensorcnt(0);
    }
    __syncthreads();

    float tv[K_TOP]; int ti[K_TOP];
#pragma unroll
    for (int i = 0; i < K_TOP; ++i) { tv[i] = NEG_INF; ti[i] = 0; }

    v8f acc = (v8f)0.0f;
    for (int t = 0; t < TCHUNKS; ++t) {
        const int p = t & 1;
        const bool more = (t + 1) < TCHUNKS;
        float4 ql[8];
        if (more) {
            const int tn  = t + 1;
            const int jn  = tn >> 3;
            const int kcn = (tn & 7) * KC;
            if (wave == 0)   // async TDM into the other buffer, overlaps compute
                tdm_load_mem_tile(mempool + (size_t)jn * 16 * C_DIM + kcn,
                                  (unsigned)(uintptr_t)&bS[p ^ 1][0][0]);
#pragma unroll
            for (int u = 0; u < 8; ++u) {        // query chunk t+1 -> registers
                int i  = tid + u * 256;
                int r4 = (i & 31) << 2;
                int c  = i >> 5;
                ql[u] = *(const float4*)(xB + (size_t)(kcn + c) * HW + r4);
            }
        }
        // ---- compute chunk t: 16x f32 WMMA
#pragma unroll
        for (int kk = 0; kk < KC; kk += 4) {
            v2f afrag = *(const v2f*)&qS[wave * 16 + mrow][kk + hf * 2];
            v2f bfrag = *(const v2f*)&bS[p][mrow][kk + hf * 2];
            acc = __builtin_amdgcn_wmma_f32_16x16x4_f32(
                      false, afrag, false, bfrag, (short)0, acc, false, false);
        }
        // ---- att tile complete every 8 chunks: spill + per-lane top-k scan
        if ((t & 7) == 7) {
            const int j = t >> 3;
#pragma unroll
            for (int v = 0; v < 8; ++v)
                attS[wave][v + hf * 8][mrow] = acc[v];
#pragma unroll
            for (int c = 0; c < 8; ++c) {
                float v = attS[wave][mrow][hf * 8 + c];
                topk_insert(tv, ti, v, j * 16 + hf * 8 + c);
            }
            acc = (v8f)0.0f;
        }
        __syncthreads();               // all waves done with qS / bS[p]
        if (more) {
#pragma unroll
            for (int u = 0; u < 8; ++u) {
                int i  = tid + u * 256;
                int r4 = (i & 31) << 2;
                int c  = i >> 5;
                qS[r4 + 0][c] = ql[u].x; qS[r4 + 1][c] = ql[u].y;
                qS[r4 + 2][c] = ql[u].z; qS[r4 + 3][c] = ql[u].w;
            }
            if (wave == 0) __builtin_amdgcn_s_wait_tensorcnt(0);
        }
        __syncthreads();               // qS + bS[p^1] ready for chunk t+1
    }

    // ---- merge the two half-lane lists for each row (wave32 shuffle)
#pragma unroll
    for (int i = 0; i < K_TOP; ++i) {
        float pv = __shfl(tv[i], lane ^ 16, 32);
        int   pi = __shfl(ti[i], lane ^ 16, 32);
        topk_insert(tv, ti, pv, pi);
    }

    // ---- softmax over top-10 (sorted descending; tv[0] is the max)
    float e[K_TOP]; float s = 0.f;
#pragma unroll
    for (int i = 0; i < K_TOP; ++i) { e[i] = __expf(tv[i] - tv[0]); s += e[i]; }
    float inv = 1.0f / s;

    const int rowG = rowBase + wave * 16 + mrow;
    if (hf == 0) {
#pragma unroll
        for (int i = 0; i < K_TOP; ++i) {
            float w = e[i] * inv;
            wS[wave][mrow][i]  = w;
            idS[wave][mrow][i] = ti[i];
            attout[(size_t)rowG * N_ITEMS + ti[i]] = w * tv[i];   // attvec * att
        }
    }

    // ---- out[row] = sum_i w_i * mempool[idx_i]  (NCHW layout by channel)
    const int hwTile = hwBase + wave * 16;
    float* outB = out + (size_t)b * C_DIM * HW + hwTile;
#pragma unroll 2
    for (int c0 = 0; c0 < C_DIM; c0 += 2) {
        int cc = c0 + hf;
        float a = 0.f;
#pragma unroll
        for (int i = 0; i < K_TOP; ++i)
            a += wS[wave][mrow][i] *
                 mempool[(size_t)idS[wave][mrow][i] * C_DIM + cc];
        outB[(size_t)cc * HW + mrow] = a;
    }
}

extern "C" void kernel_launch(void* const* d_in, const int* in_sizes, int n_in,
                              void* d_out, int out_size, void* d_ws, size_t ws_size,
                              hipStream_t stream) {
    const float* x   = (const float*)d_in[0];   // 32*512*32*32
    const float* mem = (const float*)d_in[1];   // 2000*512
    float* out    = (float*)d_out;              // first output: 16,777,216 floats
    float* attout = out + (long long)32 * 512 * 32 * 32;  // second: 32768*2000

    const long long n4 = (long long)32768 * N_ITEMS / 4;
    zero_f4<<<4096, 256, 0, stream>>>((float4*)attout, n4);
    memnet_fused<<<32768 / ROWS_WG, 256, 0, stream>>>(x, mem, out, attout);
}